// Dcls2d_5050881540607
// MI455X (gfx1250) — compile-verified
//
#include <hip/hip_runtime.h>

// ---------------- problem constants (from the reference) ----------------
#define NBATCH   32
#define CHANS    384
#define HW       56
#define PLANE    (HW * HW)          // 3136
#define NPLANES  (NBATCH * CHANS)   // 12288
#define KTAPS    36                 // 9 weights x 4 bilinear corners
#define LPAD     10                 // conv padding / kernel half-extent (21x21)

// LDS tile: 76 rows (56 + 2*10 halo), stride 80 (alignment + bank layout).
// Interior = rows [10,66) x cols [12,68): every B128 async dest is 16B aligned.
#define LROWS    76
#define LSTRIDE  80
#define LSIZE    (LROWS * LSTRIDE)  // 6080 floats = 24320 B
#define LCOL0    12
#define NHALO    2944               // halo cells = 6080 - 56*56

#define NTHREADS 224                // 7 wave32; 224 = 4 * 56 -> 14 rows/thread
#define ROWS_PT  14

// ---------------- CDNA5 async global->LDS path ----------------
#if __has_builtin(__builtin_amdgcn_global_load_async_to_lds_b128)
#define USE_ASYNC_LDS 1
#else
#define USE_ASYNC_LDS 0
#endif

typedef int v4i __attribute__((ext_vector_type(4)));
typedef v4i __attribute__((address_space(1))) gv4i_t;   // global
typedef v4i __attribute__((address_space(3))) lv4i_t;   // LDS

__device__ __forceinline__ void wait_asynccnt0() {
#if __has_builtin(__builtin_amdgcn_s_wait_asynccnt)
  __builtin_amdgcn_s_wait_asynccnt(0);
#else
  asm volatile("s_wait_asynccnt 0" ::: "memory");
#endif
}

__global__ __launch_bounds__(NTHREADS) void dcls2d_kernel(
    const float* __restrict__ x,     // (32,384,56,56)
    const float* __restrict__ w,     // (384,1,3,3)
    const float* __restrict__ P1,    // (1,3,3)
    const float* __restrict__ P2,    // (1,3,3)
    const float* __restrict__ bias,  // (384)
    float* __restrict__ out)         // (32,384,56,56)
{
  __shared__ float smem[LSIZE];
  __shared__ float s_val[KTAPS];
  __shared__ int   s_idx[KTAPS];

  const int nc = blockIdx.x;            // plane id = n*384 + c
  const int c  = nc % CHANS;
  const float* plane = x + (size_t)nc * PLANE;

  // ---- phase 1: issue async staging of the 56x56 plane into the tile
  // interior IMMEDIATELY (56 rows x 14 B128 chunks, all 16B aligned); the
  // halo zero-fill and tap setup below never touch the interior, so the
  // ASYNC writes overlap with them and only the barrier orders the readers.
  for (int t = threadIdx.x; t < PLANE / 4; t += NTHREADS) {
    int row = t / 14, seg = (t % 14) << 2;
    const float* g = plane + row * HW + seg;
    float* l = &smem[(row + LPAD) * LSTRIDE + LCOL0 + seg];
#if USE_ASYNC_LDS
    __builtin_amdgcn_global_load_async_to_lds_b128((gv4i_t*)g, (lv4i_t*)l, 0, 0);
#else
    *(float4*)l = *(const float4*)g;
#endif
  }

  // ---- phase 2: zero ONLY the halo ring (implicit conv zero-padding) ----
  // top band 10x80, bottom band 10x80, left 56x12, right 56x12 = 2944 cells.
  for (int i = threadIdx.x; i < NHALO; i += NTHREADS) {
    int idx;
    if (i < 800) {                        // rows 0..9, all cols
      idx = i;
    } else if (i < 1600) {                // rows 66..75, all cols
      idx = (66 * LSTRIDE - 800) + i;
    } else {                              // rows 10..65, cols 0..11 and 68..79
      int i3 = i - 1600;
      int r = i3 / 24, cc = i3 % 24;
      idx = (10 + r) * LSTRIDE + (cc < 12 ? cc : cc + 56);
    }
    smem[idx] = 0.0f;
  }

  // ---- tap table: mirror construct_kernel() exactly ----
  // p = clip(P + 10, 0, 20); bilinear scatter to 4 corners, clamped at border.
  if (threadIdx.x < KTAPS) {
    const int j = threadIdx.x, kk = j >> 2, corner = j & 3;
    float p1 = fminf(fmaxf(P1[kk] + 10.0f, 0.0f), 20.0f);
    float p2 = fminf(fmaxf(P2[kk] + 10.0f, 0.0f), 20.0f);
    float f1 = floorf(p1), f2 = floorf(p2);
    float r1 = p1 - f1,    r2 = p2 - f2;
    int i1 = (int)f1, i2 = (int)f2;
    int i1p = min(i1 + 1, 20), i2p = min(i2 + 1, 20);
    int a = (corner & 1) ? i1p : i1;            // kernel row 0..20
    int b = (corner & 2) ? i2p : i2;            // kernel col 0..20
    float c1 = (corner & 1) ? r1 : 1.0f - r1;
    float c2 = (corner & 2) ? r2 : 1.0f - r2;
    // out[y,x] needs in[y+a-10, x+b-10]; in LDS that's row y+a, col x+b+2.
    s_idx[j] = a * LSTRIDE + (b + 2);
    s_val[j] = w[c * 9 + kk] * (c1 * c2);       // duplicates accumulate, like .add
  }

#if USE_ASYNC_LDS
  wait_asynccnt0();                     // my async writes landed in LDS
#endif
  __syncthreads();                      // everyone's writes visible

  // ---- phase 3: each thread computes a 14-row column of outputs ----
  const int x0 = threadIdx.x % HW;
  const int y0 = (threadIdx.x / HW) * ROWS_PT;
  const float* basep = smem + y0 * LSTRIDE + x0;

  float acc[ROWS_PT];
  const float bv = bias[c];
#pragma unroll
  for (int k = 0; k < ROWS_PT; ++k) acc[k] = bv;

#pragma unroll 4
  for (int j = 0; j < KTAPS; ++j) {
    const float v = s_val[j];
    const float* p = basep + s_idx[j];          // one VALU add per tap
#pragma unroll
    for (int k = 0; k < ROWS_PT; ++k)           // ds_load imm-offset + v_fmac
      acc[k] += v * p[k * LSTRIDE];
  }

  float* o = out + (size_t)nc * PLANE + y0 * HW + x0;   // coalesced stores
#pragma unroll
  for (int k = 0; k < ROWS_PT; ++k) o[k * HW] = acc[k];
}

extern "C" void kernel_launch(void* const* d_in, const int* in_sizes, int n_in,
                              void* d_out, int out_size, void* d_ws, size_t ws_size,
                              hipStream_t stream) {
  (void)in_sizes; (void)n_in; (void)d_ws; (void)ws_size; (void)out_size;
  const float* x    = (const float*)d_in[0];
  const float* w    = (const float*)d_in[1];
  const float* P1   = (const float*)d_in[2];
  const float* P2   = (const float*)d_in[3];
  const float* bias = (const float*)d_in[4];
  float* out = (float*)d_out;

  dcls2d_kernel<<<dim3(NPLANES), dim3(NTHREADS), 0, stream>>>(x, w, P1, P2, bias, out);
}